// ReportGeneratorLSTM_36636071035280
// MI455X (gfx1250) — compile-verified
//
#include <hip/hip_runtime.h>
#include <hip/hip_bf16.h>
#include <math.h>

// ---------------- model constants ----------------
static constexpr int kV = 32000;
static constexpr int kE = 256;
static constexpr int kH = 512;
static constexpr int kB = 8;
static constexpr int kS = 192;
static constexpr int kT = 96;
static constexpr int kG = 4 * kH;     // 2048 gate width
static constexpr int kD = 2 * kH;     // 1024 bidirectional width

typedef __attribute__((ext_vector_type(2))) float v2f;
typedef __attribute__((ext_vector_type(8))) float v8f;

// ---------------------------------------------------------------------------
// Generic fp32 WMMA GEMM:  C[M,N] = A[M,K] @ Bt[N,K]^T (+ addC) (+ bias1+bias2)
// A must be physically padded to a multiple of 16 rows (padding rows only
// feed garbage into D rows that are never stored). One wave per 16x16 tile,
// 4 waves (128 threads) per block covering a 16x64 tile.
// Uses V_WMMA_F32_16X16X4_F32, accumulating over K in steps of 4.
// ---------------------------------------------------------------------------
__global__ __launch_bounds__(128)
void gemm_wmma_f32_kernel(const float* __restrict__ A, int lda,
                          const float* __restrict__ Bt, int ldb,
                          float* __restrict__ C, int ldc,
                          int M, int N, int K,
                          const float* __restrict__ addC, int ldadd,
                          const float* __restrict__ bias1,
                          const float* __restrict__ bias2)
{
    const int lane = threadIdx.x & 31;
    const int wave = threadIdx.x >> 5;
    const int hi   = lane >> 4;      // 0: K pair {0,1}, 1: K pair {2,3}
    const int l    = lane & 15;
    const int m0   = blockIdx.y << 4;
    const int n0   = (blockIdx.x * 4 + wave) << 4;
    if (n0 >= N) return;

    // A fragment: lane covers row m0+l, K slots 2*hi, 2*hi+1 (contiguous pair)
    const float* ap = A  + (size_t)(m0 + l) * lda + 2 * hi;
    // B fragment: weights are N x K row-major; lane covers col n0+l, same K pair
    const float* bp = Bt + (size_t)(n0 + l) * ldb + 2 * hi;

    v8f acc = {0.f, 0.f, 0.f, 0.f, 0.f, 0.f, 0.f, 0.f};

    int k = 0;
    for (; k + 64 <= K; k += 64) {
        // hint the next K-chunk into cache (speculative prefetch: OOB is dropped)
        __builtin_prefetch(ap + k + 64, 0, 1);
        __builtin_prefetch(bp + k + 64, 0, 1);
#pragma unroll
        for (int kk = 0; kk < 64; kk += 4) {
            v2f av; av.x = ap[k + kk]; av.y = ap[k + kk + 1];
            v2f bv; bv.x = bp[k + kk]; bv.y = bp[k + kk + 1];
            acc = __builtin_amdgcn_wmma_f32_16x16x4_f32(
                false, av, false, bv, (short)0, acc, false, false);
        }
    }
    for (; k < K; k += 4) {
        v2f av; av.x = ap[k]; av.y = ap[k + 1];
        v2f bv; bv.x = bp[k]; bv.y = bp[k + 1];
        acc = __builtin_amdgcn_wmma_f32_16x16x4_f32(
            false, av, false, bv, (short)0, acc, false, false);
    }

    const int n = n0 + l;
    float badd = 0.f;
    if (bias1) badd += bias1[n];
    if (bias2) badd += bias2[n];
#pragma unroll
    for (int r = 0; r < 8; ++r) {
        const int m = m0 + r + 8 * hi;   // C/D layout: lanes>=16 hold rows m0+8..15
        if (m < M) {
            float v = acc[r] + badd;
            if (addC) v += addC[(size_t)m * ldadd + n];
            C[(size_t)m * ldc + n] = v;
        }
    }
}

// ---------------------------------------------------------------------------
// Embedding gather: out[(t*B+b)*E + e] = emb[idx[b*idxLD + t]*E + e]
// ---------------------------------------------------------------------------
__global__ __launch_bounds__(256)
void gather_embed_kernel(const float* __restrict__ emb, const int* __restrict__ idx,
                         float* __restrict__ outp, int Tlen, int E, int idxLD)
{
    int gid = blockIdx.x * 256 + threadIdx.x;
    int total = Tlen * kB * E;
    if (gid >= total) return;
    int e = gid % E;
    int r = gid / E;
    int b = r % kB;
    int t = r / kB;
    int tok = idx[b * idxLD + t];
    outp[gid] = emb[(size_t)tok * E + e];
}

// ---------------------------------------------------------------------------
// LSTM cell pointwise update. z: 8 x 2048 (i,f,g,o blocks of 512).
// c updated in place (8x512). hpad gets new h in rows 0..7 (rows 8..15 stay 0).
// Optional extra destination ys[b*ysStride + h].
// ---------------------------------------------------------------------------
__global__ __launch_bounds__(256)
void lstm_cell_kernel(const float* __restrict__ z, float* __restrict__ c,
                      float* __restrict__ hpad, float* __restrict__ ys, int ysStride)
{
    int gid = blockIdx.x * 256 + threadIdx.x;       // < 8*512
    int b = gid >> 9;
    int h = gid & (kH - 1);
    const float* zr = z + (size_t)b * kG;
    float ig = zr[h];
    float fg = zr[kH + h];
    float gg = zr[2 * kH + h];
    float og = zr[3 * kH + h];
    ig = 1.f / (1.f + __expf(-ig));
    fg = 1.f / (1.f + __expf(-fg));
    og = 1.f / (1.f + __expf(-og));
    gg = tanhf(gg);
    float cc = fg * c[gid] + ig * gg;
    c[gid] = cc;
    float hn = og * tanhf(cc);
    hpad[gid] = hn;
    if (ys) ys[(size_t)b * ysStride + h] = hn;
}

// ---------------------------------------------------------------------------
// Attention scores: scores[b,s] = sum_h v[h] * tanh(encproj[s,b,h] + q[b,h])
// One block (128 threads = 4 waves) per (s, b).
// ---------------------------------------------------------------------------
__global__ __launch_bounds__(128)
void attn_scores_kernel(const float* __restrict__ encproj, const float* __restrict__ q,
                        const float* __restrict__ v, float* __restrict__ scores)
{
    int s = blockIdx.x, b = blockIdx.y;
    const float* ep = encproj + ((size_t)s * kB + b) * kH;
    const float* qb = q + (size_t)b * kH;
    float sum = 0.f;
    for (int h = threadIdx.x; h < kH; h += 128)
        sum += v[h] * tanhf(ep[h] + qb[h]);
    for (int off = 16; off > 0; off >>= 1)
        sum += __shfl_down(sum, off, 32);
    __shared__ float ps[4];
    if ((threadIdx.x & 31) == 0) ps[threadIdx.x >> 5] = sum;
    __syncthreads();
    if (threadIdx.x == 0)
        scores[b * kS + s] = ps[0] + ps[1] + ps[2] + ps[3];
}

// Softmax over S=192 per batch row. One block per b.
__global__ __launch_bounds__(256)
void softmax_kernel(const float* __restrict__ scores, float* __restrict__ w)
{
    int b = blockIdx.x, tid = threadIdx.x;
    __shared__ float red[256];
    float x = (tid < kS) ? scores[b * kS + tid] : -3.4e38f;
    red[tid] = x; __syncthreads();
    for (int off = 128; off > 0; off >>= 1) {
        if (tid < off) red[tid] = fmaxf(red[tid], red[tid + off]);
        __syncthreads();
    }
    float mx = red[0]; __syncthreads();
    float e = (tid < kS) ? __expf(x - mx) : 0.f;
    red[tid] = e; __syncthreads();
    for (int off = 128; off > 0; off >>= 1) {
        if (tid < off) red[tid] += red[tid + off];
        __syncthreads();
    }
    if (tid < kS) w[b * kS + tid] = e / red[0];
}

// context[b,d] = sum_s w[b,s] * enc[(s*B+b)*1024 + d]; writes rows 0..7 of ctxpad.
__global__ __launch_bounds__(256)
void context_kernel(const float* __restrict__ w, const float* __restrict__ enc,
                    float* __restrict__ ctx)
{
    int d = blockIdx.x * 256 + threadIdx.x;   // < 1024
    int b = blockIdx.y;
    float acc = 0.f;
    for (int s = 0; s < kS; ++s)
        acc += w[b * kS + s] * enc[((size_t)s * kB + b) * kD + d];
    ctx[(size_t)b * kD + d] = acc;
}

// ---------------- host helpers ----------------
static inline void gemm(hipStream_t st, const float* A, int lda, const float* Bt, int ldb,
                        float* C, int ldc, int M, int N, int K,
                        const float* addC, int ldadd, const float* b1, const float* b2)
{
    dim3 grid((N + 63) / 64, (M + 15) / 16);
    gemm_wmma_f32_kernel<<<grid, dim3(128), 0, st>>>(A, lda, Bt, ldb, C, ldc,
                                                     M, N, K, addC, ldadd, b1, b2);
}

static inline void cell(hipStream_t st, const float* z, float* c, float* hpad,
                        float* ys, int ysStride)
{
    lstm_cell_kernel<<<(kB * kH) / 256, 256, 0, st>>>(z, c, hpad, ys, ysStride);
}

extern "C" void kernel_launch(void* const* d_in, const int* in_sizes, int n_in,
                              void* d_out, int out_size, void* d_ws, size_t ws_size,
                              hipStream_t stream)
{
    (void)in_sizes; (void)n_in; (void)out_size; (void)ws_size;
    // pytree-flatten order (dict keys sorted: capitals before lowercase)
    const float* Wout   = (const float*)d_in[0];   // (V,H)
    const float* attnW  = (const float*)d_in[1];   // (H,3H)
    const float* attnB  = (const float*)d_in[2];   // (H,)
    const float* attnV  = (const float*)d_in[3];   // (H,)
    const float* bout   = (const float*)d_in[4];   // (V,)
    const float* dWhh0  = (const float*)d_in[5];   // (4H,H)
    const float* dWih0  = (const float*)d_in[6];   // (4H,E+2H)
    const float* dbhh0  = (const float*)d_in[7];
    const float* dbih0  = (const float*)d_in[8];
    const float* dWhh1  = (const float*)d_in[9];   // (4H,H)
    const float* dWih1  = (const float*)d_in[10];  // (4H,H)
    const float* dbhh1  = (const float*)d_in[11];
    const float* dbih1  = (const float*)d_in[12];
    const float* embDec = (const float*)d_in[13];  // (V,E)
    const float* embEnc = (const float*)d_in[14];  // (V,E)
    const float* eWhh[2][2]; const float* eWih[2][2];
    const float* ebhh[2][2]; const float* ebih[2][2];
    {   // enc[layer] dict {'b','f'}: d=0 -> backward, d=1 -> forward
        int p = 15;
        for (int layer = 0; layer < 2; ++layer)
            for (int d = 0; d < 2; ++d) {
                eWhh[layer][d] = (const float*)d_in[p++];
                eWih[layer][d] = (const float*)d_in[p++];
                ebhh[layer][d] = (const float*)d_in[p++];
                ebih[layer][d] = (const float*)d_in[p++];
            }
    }
    const int* src = (const int*)d_in[31];  // (B,S)
    const int* tgt = (const int*)d_in[32];  // (B,T)
    float* outp = (float*)d_out;            // (B,T,V)

    // ------- workspace carve-up (floats) -------
    float* ws = (float*)d_ws;
    size_t off = 0;
    auto take = [&](size_t n) { float* p = ws + off; off += n; return p; };
    float* xs0    = take((size_t)kS * kB * kE);   // encoder embeddings (S,B,E)
    float* gatesF = take((size_t)kS * kB * kG);   // x@Wih^T forward
    float* gatesB = take((size_t)kS * kB * kG);   // x@Wih^T backward
    float* xs1    = take((size_t)kS * kB * kD);   // layer0 output (S,B,2H)
    float* xs2    = take((size_t)kS * kB * kD);   // layer1 output = enc_out
    float* encprj = take((size_t)kS * kB * kH);   // enc_proj (S,B,H)
    float* embdec = take((size_t)kT * kB * kE);   // decoder embeddings (T,B,E)
    float* pre0   = take((size_t)kT * kB * kG);   // emb part of dec0 gates
    float* hs     = take((size_t)kT * kB * kH);   // h2 outputs, row = b*T+t
    float* hpadF  = take(16 * kH); float* cF = take(kB * kH);
    float* hpadB  = take(16 * kH); float* cB = take(kB * kH);
    float* zF     = take(kB * kG); float* zB = take(kB * kG);
    float* qbuf   = take(kB * kH);
    float* scores = take(kB * kS); float* wts = take(kB * kS);
    float* ctxpad = take(16 * kD);
    float* h1pad  = take(16 * kH); float* c1 = take(kB * kH);
    float* h2pad  = take(16 * kH); float* c2 = take(kB * kH);
    float* z0     = take(kB * kG); float* z1 = take(kB * kG);

    // ================= encoder =================
    {
        int tot = kS * kB * kE;
        gather_embed_kernel<<<(tot + 255) / 256, 256, 0, stream>>>(
            embEnc, src, xs0, kS, kE, kS);
    }
    for (int layer = 0; layer < 2; ++layer) {
        const float* Ain = layer ? xs1 : xs0;
        const int kin = layer ? kD : kE;
        float* outX = layer ? xs2 : xs1;
        // big parallel input projections for all timesteps (WMMA)
        gemm(stream, Ain, kin, eWih[layer][1], kin, gatesF, kG,
             kS * kB, kG, kin, nullptr, 0, nullptr, nullptr);
        gemm(stream, Ain, kin, eWih[layer][0], kin, gatesB, kG,
             kS * kB, kG, kin, nullptr, 0, nullptr, nullptr);
        hipMemsetAsync(hpadF, 0, 16 * kH * sizeof(float), stream);
        hipMemsetAsync(cF,    0, kB * kH * sizeof(float), stream);
        hipMemsetAsync(hpadB, 0, 16 * kH * sizeof(float), stream);
        hipMemsetAsync(cB,    0, kB * kH * sizeof(float), stream);
        for (int i = 0; i < kS; ++i) {
            const int tf = i, tb = kS - 1 - i;
            // forward dir: z = h@Whh^T + gates[t] + bih + bhh
            gemm(stream, hpadF, kH, eWhh[layer][1], kH, zF, kG, kB, kG, kH,
                 gatesF + (size_t)tf * kB * kG, kG, ebih[layer][1], ebhh[layer][1]);
            cell(stream, zF, cF, hpadF, outX + (size_t)tf * kB * kD, kD);
            // backward dir
            gemm(stream, hpadB, kH, eWhh[layer][0], kH, zB, kG, kB, kG, kH,
                 gatesB + (size_t)tb * kB * kG, kG, ebih[layer][0], ebhh[layer][0]);
            cell(stream, zB, cB, hpadB, outX + (size_t)tb * kB * kD + kH, kD);
        }
    }

    // enc_proj = enc_out @ We^T + attn_b   (We = attn_W[:, :2H], ldb = 3H)
    gemm(stream, xs2, kD, attnW, 3 * kH, encprj, kH,
         kS * kB, kH, kD, nullptr, 0, attnB, nullptr);

    // ================= decoder =================
    {
        int tot = kT * kB * kE;
        gather_embed_kernel<<<(tot + 255) / 256, 256, 0, stream>>>(
            embDec, tgt, embdec, kT, kE, kT);
    }
    // embedding part of dec0 gates: pre0 = emb_t @ Wih0[:, :E]^T
    gemm(stream, embdec, kE, dWih0, kE + kD, pre0, kG,
         kT * kB, kG, kE, nullptr, 0, nullptr, nullptr);
    hipMemsetAsync(h1pad, 0, 16 * kH * sizeof(float), stream);
    hipMemsetAsync(c1,    0, kB * kH * sizeof(float), stream);
    hipMemsetAsync(h2pad, 0, 16 * kH * sizeof(float), stream);
    hipMemsetAsync(c2,    0, kB * kH * sizeof(float), stream);
    hipMemsetAsync(ctxpad, 0, 16 * kD * sizeof(float), stream);

    for (int t = 0; t < kT; ++t) {
        // q = h2 @ Wh^T   (Wh = attn_W[:, 2H:], ldb = 3H)
        gemm(stream, h2pad, kH, attnW + kD, 3 * kH, qbuf, kH,
             kB, kH, kH, nullptr, 0, nullptr, nullptr);
        attn_scores_kernel<<<dim3(kS, kB), 128, 0, stream>>>(encprj, qbuf, attnV, scores);
        softmax_kernel<<<kB, 256, 0, stream>>>(scores, wts);
        context_kernel<<<dim3(kD / 256, kB), 256, 0, stream>>>(wts, xs2, ctxpad);
        // dec LSTM0: z0 = ctx @ Wih0[:, E:]^T + pre0[t]; z0 += h1 @ Whh0^T + biases
        gemm(stream, ctxpad, kD, dWih0 + kE, kE + kD, z0, kG, kB, kG, kD,
             pre0 + (size_t)t * kB * kG, kG, nullptr, nullptr);
        gemm(stream, h1pad, kH, dWhh0, kH, z0, kG, kB, kG, kH,
             z0, kG, dbih0, dbhh0);
        cell(stream, z0, c1, h1pad, nullptr, 0);
        // dec LSTM1: z1 = h1 @ Wih1^T; z1 += h2 @ Whh1^T + biases
        gemm(stream, h1pad, kH, dWih1, kH, z1, kG, kB, kG, kH,
             nullptr, 0, nullptr, nullptr);
        gemm(stream, h2pad, kH, dWhh1, kH, z1, kG, kB, kG, kH,
             z1, kG, dbih1, dbhh1);
        cell(stream, z1, c2, h2pad, hs + (size_t)t * kH, kT * kH);  // hs[b*T+t]
    }

    // logits (B,T,V): rows b*T+t of hs @ Wout^T + bout  -> dominant WMMA GEMM
    gemm(stream, hs, kH, Wout, kH, outp, kV,
         kT * kB, kV, kH, nullptr, 0, bout, nullptr);
}